// kernel_generated_0_21887153341251
// MI455X (gfx1250) — compile-verified
//
#include <hip/hip_runtime.h>

// fp32 WMMA fragments on gfx1250 (wave32):
//   A (16x4 f32): 2 VGPRs/lane ; B (4x16 f32): 2 VGPRs/lane ; C/D (16x16 f32): 8 VGPRs/lane
typedef __attribute__((ext_vector_type(2))) float v2f;
typedef __attribute__((ext_vector_type(8))) float v8f;

#define HW    12544   // 112*112
#define KDIM  84      // 12 in_ch * 7 taps
#define NPAD  16      // out-channel dim padded 12 -> 16 with zero weights
#define AROW  118     // im2col row stride: 112 + 6 halo columns
#define AOFS  3       // WMMA reads row cells [3, 115)

// One block per (group g, height row h). 7 waves; wave t computes the 16-wide
// width tile [16t, 16t+16) of all 12 output channels via 21 x WMMA_F32_16X16X4_F32.
// Staging is scatter-style: every global load and every LDS store is
// unconditional with compile-time immediate offsets (no EXEC juggling).
__global__ __launch_bounds__(224) void conv_w7_g2_roll_wmma(
    const float* __restrict__ x,   // (1,24,112,112)
    const float* __restrict__ W,   // (12 in_c, 12 out_c, 7 tap)
    float* __restrict__ out)       // (1,24,112,112), rolled +1 along height
{
    __shared__ float a2[KDIM * AROW];   // im2col rows with halo: A[kk][0..117]
    __shared__ float b2[KDIM * NPAD];   // weights K-major: B[kk][n], N padded

    const int tid   = threadIdx.x;
    const int g     = blockIdx.x / 112;
    const int h     = blockIdx.x % 112;
    const int lane  = tid & 31;
    const int wtile = tid >> 5;         // 0..6 : which 16-wide M tile

    if (tid < 112) {
        // ---- Phase A: scatter im2col. Thread owns source column wsrc. ----
        // a2[(i*7+j)*AROW + wsrc + 6 - j] = x[i][wsrc]; all indices in [0,117].
        const int wsrc = tid;
        const float* __restrict__ xb = x + (g * 12) * HW + h * 112 + wsrc;
        float* __restrict__ ab = &a2[wsrc];
        #pragma unroll
        for (int i = 0; i < 12; ++i) {
            const float v = xb[i * HW];          // always in-bounds, imm offset
            #pragma unroll
            for (int j = 0; j < 7; ++j)
                ab[(i * 7 + j) * AROW + (6 - j)] = v;   // imm offsets, one base
        }
    } else {
        // ---- Phase B: weights. t2 in [0,112): n = t2&15, tap j = t2>>4. ----
        // kk = i*7 + j with i = rep (division-free); N >= 12 zero-padded.
        const int t2 = tid - 112;
        const int n  = t2 & 15;
        const int j  = t2 >> 4;                  // 0..6
        const int nc = (n < 12) ? n : 11;        // clamp for safe global read
        const float* __restrict__ wb = W + nc * 7 + j;
        float* __restrict__ bb = &b2[j * NPAD + n];
        #pragma unroll
        for (int i = 0; i < 12; ++i) {
            const float v = wb[i * 84];          // imm offset
            bb[i * 7 * NPAD] = (n < 12) ? v : 0.0f;
        }
    }

    // ---- Halo zeroing: the 144 im2col cells whose source is outside [0,112).
    // Per row j: front cells [3, 6-j) (j<3), back cells (117-j, 114] (j>3).
    // (j, cell) decoded from nibble-packed constants; one store per thread.
    if (tid < 144) {
        const int i = tid / 12;
        const int q = tid % 12;
        const int j = (int)((0x666554211000ULL >> (4 * q)) & 0xF);
        const int d = (int)((0x210212010210ULL >> (4 * q)) & 0xF);
        const int cell = ((q < 6) ? 3 : 112) + d;
        a2[(i * 7 + j) * AROW + cell] = 0.0f;
    }
    __syncthreads();

    const int m     = lane & 15;   // A: M row in tile ; B/C/D: N column
    const int khalf = lane >> 4;   // 0 -> K+{0,1} ; 1 -> K+{2,3}

    // Bases computed once; every loop access is base + compile-time offset.
    const float* __restrict__ abase = &a2[khalf * 2 * AROW + AOFS + wtile * 16 + m];
    const float* __restrict__ bbase = &b2[khalf * 2 * NPAD + m];

    v8f acc = {};

    #pragma unroll
    for (int kc = 0; kc < KDIM; kc += 4) {
        v2f a, b;
        a.x = abase[(kc + 0) * AROW];
        a.y = abase[(kc + 1) * AROW];
        b.x = bbase[(kc + 0) * NPAD];
        b.y = bbase[(kc + 1) * NPAD];
        // (neg_a, A, neg_b, B, c_mod, C, reuse_a, reuse_b)
        acc = __builtin_amdgcn_wmma_f32_16x16x4_f32(
                  false, a, false, b, (short)0, acc, false, false);
    }

    // roll(+1, axis=height): row h results are written to row (h+1) % 112.
    const int hout = (h + 1) % 112;
    if (m < 12) {   // n = m ; drop padded output channels
        #pragma unroll
        for (int r = 0; r < 8; ++r) {
            const int mm = r + 8 * khalf;  // D row = vgpr + 8*(lane>=16)
            out[(g * 12 + m) * HW + hout * 112 + wtile * 16 + mm] = acc[r];
        }
    }
}

extern "C" void kernel_launch(void* const* d_in, const int* in_sizes, int n_in,
                              void* d_out, int out_size, void* d_ws, size_t ws_size,
                              hipStream_t stream) {
    const float* x = (const float*)d_in[0];   // 1*24*112*112 fp32
    const float* W = (const float*)d_in[1];   // 12*12*7 fp32
    float* out     = (float*)d_out;           // 1*24*112*112 fp32

    conv_w7_g2_roll_wmma<<<dim3(2 * 112), dim3(224), 0, stream>>>(x, W, out);
}